// Voxelizer_58488864637209
// MI455X (gfx1250) — compile-verified
//
#include <hip/hip_runtime.h>

// ---------------------------------------------------------------------------
// Voxelizer: grid = zeros(4,50,800,800); grid[b, bz, by, bx] max= mask(1.0)
// Pure HBM-bound: 512MB fill + tiny scatter. No matmul structure -> no WMMA;
// optimize the data-movement path instead (async LDS->global b128 / NT b128).
// ---------------------------------------------------------------------------

typedef float v4f __attribute__((ext_vector_type(4)));
// Match the builtin's parameter pointee type exactly (from hipcc diagnostic):
// '__attribute__((__vector_size__(4 * sizeof(int)))) int'
typedef int v4i __attribute__((vector_size(4 * sizeof(int))));
typedef __attribute__((address_space(1))) v4i* vx_gptr_t;  // global (device)
typedef __attribute__((address_space(3))) v4i* vx_lptr_t;  // LDS

#define VX_XMIN  (-80.0f)
#define VX_XMAX  ( 80.0f)
#define VX_YMIN  (-80.0f)
#define VX_YMAX  ( 80.0f)
#define VX_ZMIN  (  0.0f)
#define VX_STEP  (  0.2f)
#define VX_D 50
#define VX_H 800
#define VX_W 800

#if __has_builtin(__builtin_amdgcn_global_store_async_from_lds_b128)
#define VX_ASYNC_FILL 1
#else
#define VX_ASYNC_FILL 0
#endif

// ---------------------------------------------------------------------------
// Phase 1: zero 512MB. One 16B store per lane per iteration = 512B/wave32
// per store instruction. Async-from-LDS (ASYNCcnt) or NT-hinted b128 stores.
// ---------------------------------------------------------------------------
__global__ __launch_bounds__(256) void vox_fill_kernel(v4f* __restrict__ out,
                                                       long long n4) {
  long long tid    = (long long)blockIdx.x * blockDim.x + threadIdx.x;
  long long stride = (long long)gridDim.x * blockDim.x;

#if VX_ASYNC_FILL
  // CDNA5 async path: stage 16B of zeros per lane in LDS once, then stream
  // GLOBAL_STORE_ASYNC_FROM_LDS_B128 (tracked by ASYNCcnt) to global memory.
  __shared__ __align__(16) v4f zbuf[256];
  zbuf[threadIdx.x] = (v4f){0.0f, 0.0f, 0.0f, 0.0f};
  __syncthreads();
  vx_lptr_t lds_src = (vx_lptr_t)&zbuf[threadIdx.x];
  for (long long i = tid; i < n4; i += stride) {
    __builtin_amdgcn_global_store_async_from_lds_b128(
        (vx_gptr_t)&out[i], lds_src, /*offset=*/0, /*cpol=*/0);
  }
#if __has_builtin(__builtin_amdgcn_s_wait_asynccnt)
  __builtin_amdgcn_s_wait_asynccnt(0);
#else
  asm volatile("s_wait_asynccnt 0" ::: "memory");
#endif
#else
  // Fallback: plain wide stores with non-temporal hint (TH=NT).
  v4f z = {0.0f, 0.0f, 0.0f, 0.0f};
  for (long long i = tid; i < n4; i += stride) {
    __builtin_nontemporal_store(z, &out[i]);
  }
#endif
}

// ---------------------------------------------------------------------------
// Phase 2: scatter. One thread per point. All surviving writers store the
// identical value 1.0f, so races are benign and no atomics are needed.
// Division by STEP (not reciprocal-multiply) to match the f32 reference.
// ---------------------------------------------------------------------------
__global__ __launch_bounds__(256) void vox_scatter_kernel(
    const float* __restrict__ pts, float* __restrict__ grid, int total,
    int n_per_batch) {
  int i = blockIdx.x * blockDim.x + threadIdx.x;
  if (i >= total) return;

  const float* p = pts + (size_t)i * 3;
  // gfx1250 prefetch (global_prefetch_b8) a block's worth of points ahead;
  // speculative, so an OOB tail address is silently dropped.
  __builtin_prefetch(p + (size_t)3 * 256, 0, 0);

  float x = p[0];
  float y = p[1];
  float z = p[2];

  // x/y masked if out of bounds (z is only clamped, per the reference).
  if (x < VX_XMIN || x > VX_XMAX || y < VX_YMIN || y > VX_YMAX) return;

  int bx = (int)floorf((x - VX_XMIN) / VX_STEP);
  int by = (int)floorf((VX_YMAX - y) / VX_STEP);
  int bz = (int)floorf((z - VX_ZMIN) / VX_STEP);
  bx = min(max(bx, 0), VX_W - 1);
  by = min(max(by, 0), VX_H - 1);
  bz = min(max(bz, 0), VX_D - 1);

  int b = i / n_per_batch;
  size_t off = (((size_t)b * VX_D + bz) * VX_H + by) * VX_W + bx;
  __builtin_nontemporal_store(1.0f, &grid[off]);
}

// ---------------------------------------------------------------------------
extern "C" void kernel_launch(void* const* d_in, const int* in_sizes, int n_in,
                              void* d_out, int out_size, void* d_ws,
                              size_t ws_size, hipStream_t stream) {
  (void)n_in; (void)d_ws; (void)ws_size;

  const float* pts = (const float*)d_in[0];   // [B, N, 3] float32
  float* grid      = (float*)d_out;           // [B, D, H, W] float32

  // 128M floats, exactly divisible by 4 -> float4 count.
  long long n4 = (long long)out_size / 4;

  int total       = in_sizes[0] / 3;  // B*N = 800000
  int n_per_batch = total / 4;        // N   = 200000 (B fixed at 4)

  // Fill: 4096 blocks x 256 threads -> ~32 b128 stores per thread.
  vox_fill_kernel<<<4096, 256, 0, stream>>>((v4f*)grid, n4);

  // Scatter: one thread per point.
  int blocks = (total + 255) / 256;
  vox_scatter_kernel<<<blocks, 256, 0, stream>>>(pts, grid, total, n_per_batch);
}